// SelfAttention_45140106281268
// MI455X (gfx1250) — compile-verified
//
#include <hip/hip_runtime.h>

// MI455X / gfx1250, wave32. Fused self-attention:
//   qkv_kernel  : q,k,v 1x1-conv GEMMs -> d_ws
//   attn_kernel : energy (WMMA f32 16x16x4) -> LDS panel -> softmax
//                 -> async LDS->HBM store of attention (overlaps with...)
//                 -> V*attn^T (WMMA f32, K=4096) -> attn_out, gamma*attn_out
//   ln_kernel   : residual + channel LayerNorm -> out

typedef __attribute__((ext_vector_type(2))) float v2f;
typedef __attribute__((ext_vector_type(8))) float v8f;
typedef __attribute__((ext_vector_type(4))) int   v4i;

#define BATCH 8
#define CCH   64
#define AD    8
#define NPX   4096
#define LDP   4100   // padded LDS row stride: 16B-aligned rows (B128 async), 4100%64==4 -> conflict-free transposed reads

#define AS1 __attribute__((address_space(1)))
#define AS3 __attribute__((address_space(3)))

#if defined(__gfx1250__) && __has_builtin(__builtin_amdgcn_global_store_async_from_lds_b128)
#define HAVE_ASYNC_ST 1
#else
#define HAVE_ASYNC_ST 0
#endif

// ---------------------------------------------------------------------------
// Kernel 1: q = Wq*x + bq, k = Wk*x + bk, v = Wv*x + bv   (per-pixel, K = C = 64)
// ---------------------------------------------------------------------------
__global__ __launch_bounds__(256) void qkv_kernel(
    const float* __restrict__ x,
    const float* __restrict__ Wq, const float* __restrict__ bq,
    const float* __restrict__ Wk, const float* __restrict__ bk,
    const float* __restrict__ Wv, const float* __restrict__ bv,
    float* __restrict__ qo, float* __restrict__ ko, float* __restrict__ vo)
{
    int gid = blockIdx.x * 256 + threadIdx.x;
    int n   = gid & (NPX - 1);
    int row = (gid >> 12) % 80;
    int b   = gid / (NPX * 80);

    const float* Wrow;
    float bias;
    float* dst;
    if (row < AD)        { Wrow = Wq + row * CCH;       bias = bq[row];       dst = qo + ((size_t)b * AD  + row) * NPX; }
    else if (row < 2*AD) { int r = row - AD;   Wrow = Wk + r * CCH; bias = bk[r]; dst = ko + ((size_t)b * AD  + r) * NPX; }
    else                 { int r = row - 2*AD; Wrow = Wv + r * CCH; bias = bv[r]; dst = vo + ((size_t)b * CCH + r) * NPX; }

    const float* xb = x + (size_t)b * CCH * NPX + n;
    float sum = bias;
    #pragma unroll
    for (int c = 0; c < CCH; ++c)
        sum += Wrow[c] * xb[(size_t)c * NPX];
    dst[n] = sum;
}

// ---------------------------------------------------------------------------
// Kernel 2: fused energy -> softmax -> async attention writeback -> V*attn^T
// One workgroup (8 wave32) owns 16 query rows of one batch.
// Dynamic LDS panel: sAttn[16][LDP] = 262,400 B (WGP LDS = 320 KB).
// ---------------------------------------------------------------------------
extern __shared__ float sAttn[];   // [16][LDP]
#define SA(r, c) sAttn[(size_t)(r) * LDP + (c)]

__global__ __launch_bounds__(256) void attn_kernel(
    const float* __restrict__ q, const float* __restrict__ k, const float* __restrict__ v,
    const float* __restrict__ gamma_p,
    float* __restrict__ attn_g,    // [B][N][N]
    float* __restrict__ fdelta_g,  // [B][C][N]
    float* __restrict__ attnv_g)   // [B][C][N]
{
    __shared__ float sRed[4][32][8];   // cross-wave K-half reduction

    const int n0    = blockIdx.x * 16;
    const int b     = blockIdx.y;
    const int t     = threadIdx.x;
    const int lane  = t & 31;
    const int w     = t >> 5;              // wave 0..7
    const int lm    = lane & 15;
    const int khalf = (lane < 16) ? 0 : 2; // ISA A/B lane layout: low half K{0,1}, high half K{2,3}

    const float* qb = q + (size_t)b * AD  * NPX;
    const float* kb = k + (size_t)b * AD  * NPX;
    const float* vb = v + (size_t)b * CCH * NPX;

    // ---- Phase 1: E[16][4096] = Q(16x8) * K(8x4096), V_WMMA_F32_16X16X4_F32, 2 chained per tile
    v2f a0, a1;                                   // A 16x4 frags (K 0..3 and 4..7)
    a0.x = qb[(size_t)(khalf + 0) * NPX + n0 + lm];
    a0.y = qb[(size_t)(khalf + 1) * NPX + n0 + lm];
    a1.x = qb[(size_t)(khalf + 4) * NPX + n0 + lm];
    a1.y = qb[(size_t)(khalf + 5) * NPX + n0 + lm];

    const int mrow = (lane < 16) ? 0 : 8;         // D rows r (low lanes) / r+8 (high lanes)
    for (int mt = w * 32; mt < w * 32 + 32; ++mt) {
        const int m0 = mt * 16;
        v2f b0, b1;                               // B 4x16 frags
        b0.x = kb[(size_t)(khalf + 0) * NPX + m0 + lm];
        b0.y = kb[(size_t)(khalf + 1) * NPX + m0 + lm];
        b1.x = kb[(size_t)(khalf + 4) * NPX + m0 + lm];
        b1.y = kb[(size_t)(khalf + 5) * NPX + m0 + lm];
        v8f d = {};
        d = __builtin_amdgcn_wmma_f32_16x16x4_f32(false, a0, false, b0, (short)0, d, false, false);
        d = __builtin_amdgcn_wmma_f32_16x16x4_f32(false, a1, false, b1, (short)0, d, false, false);
        #pragma unroll
        for (int r = 0; r < 8; ++r)
            SA(mrow + r, m0 + lm) = d[r];
    }
    __syncthreads();

    // ---- Phase 2: row softmax fully in LDS (normalized panel kept resident)
    for (int r = w * 2; r < w * 2 + 2; ++r) {
        float mx = -3.402823466e38f;
        for (int j = lane; j < NPX; j += 32) mx = fmaxf(mx, SA(r, j));
        #pragma unroll
        for (int off = 16; off; off >>= 1) mx = fmaxf(mx, __shfl_xor(mx, off, 32));
        float s = 0.f;
        for (int j = lane; j < NPX; j += 32) {
            float e = __expf(SA(r, j) - mx);
            SA(r, j) = e;
            s += e;
        }
        #pragma unroll
        for (int off = 16; off; off >>= 1) s += __shfl_xor(s, off, 32);
        const float inv = 1.f / s;
        for (int j = lane; j < NPX; j += 32)
            SA(r, j) *= inv;
    }
    __syncthreads();

    // ---- Attention writeback: 16 x 4096 fp32 panel, LDS -> HBM.
    // Async DMA (GLOBAL_STORE_ASYNC_FROM_LDS_B128, ASYNCcnt) so the store stream
    // drains concurrently with the phase-3 WMMA loop (both only READ sAttn).
    {
        float* abase = attn_g + ((size_t)b * NPX + n0) * NPX;
        #pragma unroll 4
        for (int i = 0; i < 64; ++i) {
            const int s   = t + 256 * i;       // 16384 segments of 4 floats
            const int r   = s >> 10;
            const int col = (s & 1023) << 2;
            float* gdst = abase + (size_t)r * NPX + col;
#if HAVE_ASYNC_ST
            __builtin_amdgcn_global_store_async_from_lds_b128(
                (AS1 v4i*)(unsigned long long)(uintptr_t)gdst,
                (AS3 v4i*)(unsigned int)(uintptr_t)&SA(r, col),
                0, 0);
#else
            float4 vv = *(const float4*)&SA(r, col);
            *(float4*)gdst = vv;
#endif
        }
    }

    // ---- Phase 3: O[64][16] = V(64x4096) * attn^T(4096x16); c-tile = w&3, K-half = w>>2
    const int ct = w & 3, c0 = ct * 16, kh = w >> 2, mbase = kh * 2048;
    const v8f zero = {};
    v8f acc[4] = {zero, zero, zero, zero};        // 4 accumulators break the WMMA RAW chain
    const float* vrow = vb + (size_t)(c0 + lm) * NPX;
    for (int mm = mbase; mm < mbase + 2048; mm += 16) {
        #pragma unroll
        for (int u = 0; u < 4; ++u) {
            const int m0 = mm + u * 4;
            v2f av;  av.x  = vrow[m0 + khalf];    av.y  = vrow[m0 + khalf + 1];
            v2f bv2; bv2.x = SA(lm, m0 + khalf);  bv2.y = SA(lm, m0 + khalf + 1); // transposed, pad -> no conflict
            acc[u] = __builtin_amdgcn_wmma_f32_16x16x4_f32(false, av, false, bv2, (short)0, acc[u], false, false);
        }
    }
    v8f tot;
    #pragma unroll
    for (int r = 0; r < 8; ++r) tot[r] = (acc[0][r] + acc[1][r]) + (acc[2][r] + acc[3][r]);

    if (kh == 1) {
        #pragma unroll
        for (int r = 0; r < 8; ++r) sRed[ct][lane][r] = tot[r];
    }
    __syncthreads();
    if (kh == 0) {
        const float gamma = gamma_p[0];
        const int crow = c0 + ((lane < 16) ? 0 : 8);
        #pragma unroll
        for (int r = 0; r < 8; ++r) {
            float o = tot[r] + sRed[ct][lane][r];
            size_t idx = ((size_t)b * CCH + (crow + r)) * NPX + n0 + lm;
            attnv_g[idx]  = o;
            fdelta_g[idx] = gamma * o;
        }
    }

#if HAVE_ASYNC_ST
#if __has_builtin(__builtin_amdgcn_s_wait_asynccnt)
    __builtin_amdgcn_s_wait_asynccnt(0);
#else
    asm volatile("s_wait_asynccnt 0" ::: "memory");
#endif
#endif
}

// ---------------------------------------------------------------------------
// Kernel 3: pre = x + feature_delta; LayerNorm over C=64 (channels-last), biased var.
// ---------------------------------------------------------------------------
__global__ __launch_bounds__(256) void ln_kernel(
    const float* __restrict__ x, const float* __restrict__ fdelta,
    const float* __restrict__ ln_w, const float* __restrict__ ln_b,
    float* __restrict__ out)
{
    __shared__ float s1[4][64], s2[4][64], sMu[64], sRs[64];
    const int blocksPerB = NPX / 64;
    const int b  = blockIdx.x / blocksPerB;
    const int p0 = (blockIdx.x % blocksPerB) * 64;
    const int tx = threadIdx.x & 63, ty = threadIdx.x >> 6;

    const float* xb = x      + (size_t)b * CCH * NPX;
    const float* fb = fdelta + (size_t)b * CCH * NPX;

    float a1 = 0.f, a2 = 0.f;
    #pragma unroll
    for (int i = 0; i < 16; ++i) {
        const int c = ty * 16 + i;
        float vv = xb[(size_t)c * NPX + p0 + tx] + fb[(size_t)c * NPX + p0 + tx];
        a1 += vv; a2 += vv * vv;
    }
    s1[ty][tx] = a1; s2[ty][tx] = a2;
    __syncthreads();
    if (ty == 0) {
        float t1 = s1[0][tx] + s1[1][tx] + s1[2][tx] + s1[3][tx];
        float t2 = s2[0][tx] + s2[1][tx] + s2[2][tx] + s2[3][tx];
        float mu  = t1 * (1.f / 64.f);
        float var = t2 * (1.f / 64.f) - mu * mu;
        sMu[tx] = mu;
        sRs[tx] = rsqrtf(var + 1e-5f);
    }
    __syncthreads();
    const float mu = sMu[tx], rs = sRs[tx];
    #pragma unroll
    for (int i = 0; i < 16; ++i) {
        const int c = ty * 16 + i;
        size_t idx = (size_t)c * NPX + p0 + tx;
        float vv = xb[idx] + fb[idx];
        out[(size_t)b * CCH * NPX + idx] = (vv - mu) * rs * ln_w[c] + ln_b[c];
    }
}

// ---------------------------------------------------------------------------
extern "C" void kernel_launch(void* const* d_in, const int* in_sizes, int n_in,
                              void* d_out, int out_size, void* d_ws, size_t ws_size,
                              hipStream_t stream)
{
    (void)in_sizes; (void)n_in; (void)out_size; (void)ws_size;

    const float* x     = (const float*)d_in[0];
    const float* Wq    = (const float*)d_in[1];
    const float* bq    = (const float*)d_in[2];
    const float* Wk    = (const float*)d_in[3];
    const float* bk    = (const float*)d_in[4];
    const float* Wv    = (const float*)d_in[5];
    const float* bv    = (const float*)d_in[6];
    const float* gamma = (const float*)d_in[7];
    const float* ln_w  = (const float*)d_in[8];
    const float* ln_b  = (const float*)d_in[9];

    // d_out: [out | attention | feature_delta | attn_out] flat fp32
    float* out    = (float*)d_out;
    float* attn   = out    + (size_t)BATCH * CCH * NPX;
    float* fdelta = attn   + (size_t)BATCH * NPX * NPX;
    float* attnv  = fdelta + (size_t)BATCH * CCH * NPX;

    // d_ws: q[B][A][N] | k[B][A][N] | v[B][C][N]  (10.5 MB)
    float* q  = (float*)d_ws;
    float* kk = q  + (size_t)BATCH * AD * NPX;
    float* v  = kk + (size_t)BATCH * AD * NPX;

    qkv_kernel<<<BATCH * 80 * NPX / 256, 256, 0, stream>>>(x, Wq, bq, Wk, bk, Wv, bv, q, kk, v);

    dim3 g2(NPX / 16, BATCH);
    size_t smem = (size_t)16 * LDP * sizeof(float);   // 262,400 B LDS panel
    attn_kernel<<<g2, 256, smem, stream>>>(q, kk, v, gamma, attn, fdelta, attnv);

    ln_kernel<<<BATCH * (NPX / 64), 256, 0, stream>>>(x, fdelta, ln_w, ln_b, out);
}